// Attention_47665547051353
// MI455X (gfx1250) — compile-verified
//
#include <hip/hip_runtime.h>

// Sizes fixed by the reference.
#define BB 4
#define CC 512
#define NN 2048
#define HH 8
#define HD 64
#define SCALE_F 0.125f   // 64^-0.5

typedef __attribute__((ext_vector_type(2))) float v2f;
typedef __attribute__((ext_vector_type(8))) float v8f;

__device__ __forceinline__ v8f wmma_f32(v2f a, v2f b, v8f c) {
  // D = A(16x4,f32) * B(4x16,f32) + C(16x16,f32)
  return __builtin_amdgcn_wmma_f32_16x16x4_f32(false, a, false, b,
                                               (short)0, c, false, false);
}

__device__ __forceinline__ v8f zero8() {
  v8f z;
#pragma unroll
  for (int i = 0; i < 8; ++i) z[i] = 0.0f;
  return z;
}

// ---------------------------------------------------------------------------
// Kernel 1 & 4: Y[b,m,n] = sum_c W[m,c] * X[b,c,n] (+ bias[m])
// One wave (32 threads) computes a 32(M) x 64(N) tile: 2x4 WMMA accumulators.
// ---------------------------------------------------------------------------
__global__ void k_gemm_wn(const float* __restrict__ W,   // [M][CC]
                          const float* __restrict__ X,   // [BB][CC][NN]
                          float* __restrict__ Y,         // [BB][M][NN]
                          const float* __restrict__ bias,
                          int M) {
  const int lane = threadIdx.x;
  const int nlo  = lane & 15;
  const int half = lane >> 4;
  const int kk   = half * 2;
  const int n0   = blockIdx.x * 64;
  const int m0   = blockIdx.y * 32;
  const int b    = blockIdx.z;
  const float* Xb = X + (size_t)b * CC * NN;

  v8f acc[2][4];
#pragma unroll
  for (int mt = 0; mt < 2; ++mt)
#pragma unroll
    for (int nt = 0; nt < 4; ++nt) acc[mt][nt] = zero8();

  for (int k = 0; k < CC; k += 4) {
    v2f a[2];
#pragma unroll
    for (int mt = 0; mt < 2; ++mt) {
      const float* wp = W + (size_t)(m0 + mt * 16 + nlo) * CC + k + kk;
      a[mt].x = wp[0];
      a[mt].y = wp[1];
    }
#pragma unroll
    for (int nt = 0; nt < 4; ++nt) {
      const float* xp = Xb + (size_t)(k + kk) * NN + n0 + nt * 16 + nlo;
      v2f bv;
      bv.x = xp[0];
      bv.y = xp[NN];
#pragma unroll
      for (int mt = 0; mt < 2; ++mt)
        acc[mt][nt] = wmma_f32(a[mt], bv, acc[mt][nt]);
    }
  }

#pragma unroll
  for (int mt = 0; mt < 2; ++mt)
#pragma unroll
    for (int v = 0; v < 8; ++v) {
      const int row = m0 + mt * 16 + v + 8 * half;
      const float bval = bias ? bias[row] : 0.0f;
      float* yp = Y + ((size_t)b * M + row) * NN + n0 + nlo;
#pragma unroll
      for (int nt = 0; nt < 4; ++nt) yp[nt * 16] = acc[mt][nt][v] + bval;
    }
}

// ---------------------------------------------------------------------------
// Kernel 2: per-key-row softmax stats. s[i,j] = SCALE * sum_d k[d,i] q[d,j].
// One wave handles 16 key rows (i0..i0+15) and streams all j in tiles of 64,
// maintaining an online max / rescaled exp-sum. Writes m_i and 1/Z_i.
// ---------------------------------------------------------------------------
__global__ void k_attn_stats(const float* __restrict__ qkv,   // [BB][3*CC][NN]
                             float* __restrict__ rowmax,      // [BB][HH][NN]
                             float* __restrict__ rowinv) {    // [BB][HH][NN]
  const int lane = threadIdx.x;
  const int nlo  = lane & 15;
  const int half = lane >> 4;
  const int kk   = half * 2;
  const int i0   = blockIdx.x * 16;
  const int h    = blockIdx.y;
  const int b    = blockIdx.z;

  const float* base = qkv + ((size_t)b * (3 * CC) + (size_t)h * (3 * HD)) * NN;
  const float* qp = base;                        // q: rows [0,64)
  const float* kp = base + (size_t)HD * NN;      // k: rows [64,128)

  // Preload A = K^T rows i0..i0+15 (16 k-steps of the d=0..63 contraction).
  v2f areg[16];
#pragma unroll
  for (int ks = 0; ks < 16; ++ks) {
    const float* p = kp + (size_t)(4 * ks + kk) * NN + i0 + nlo;
    areg[ks].x = p[0];
    areg[ks].y = p[NN];
  }

  float runm[8], psum[8];
#pragma unroll
  for (int v = 0; v < 8; ++v) { runm[v] = -3.0e38f; psum[v] = 0.0f; }

  for (int j0 = 0; j0 < NN; j0 += 64) {
    v8f s[4];
#pragma unroll
    for (int t = 0; t < 4; ++t) s[t] = zero8();
#pragma unroll
    for (int ks = 0; ks < 16; ++ks) {
#pragma unroll
      for (int t = 0; t < 4; ++t) {
        const float* p = qp + (size_t)(4 * ks + kk) * NN + j0 + t * 16 + nlo;
        v2f bv;
        bv.x = p[0];
        bv.y = p[NN];
        s[t] = wmma_f32(areg[ks], bv, s[t]);
      }
    }
    // Online softmax update; row of element v = i0 + v + 8*half, cols across
    // the 16 lanes of this half -> reduce with width-16 xor shuffles.
#pragma unroll
    for (int v = 0; v < 8; ++v) {
      float lm = fmaxf(fmaxf(s[0][v], s[1][v]), fmaxf(s[2][v], s[3][v])) * SCALE_F;
#pragma unroll
      for (int o = 1; o < 16; o <<= 1) lm = fmaxf(lm, __shfl_xor(lm, o, 16));
      const float newm = fmaxf(runm[v], lm);
      const float f = __expf(runm[v] - newm);
      float a = 0.0f;
#pragma unroll
      for (int t = 0; t < 4; ++t) a += __expf(s[t][v] * SCALE_F - newm);
      psum[v] = psum[v] * f + a;
      runm[v] = newm;
    }
  }

#pragma unroll
  for (int v = 0; v < 8; ++v) {
    float ps = psum[v];
#pragma unroll
    for (int o = 1; o < 16; o <<= 1) ps += __shfl_xor(ps, o, 16);
    if (nlo == 0) {
      const int row = i0 + v + 8 * half;
      const size_t idx = ((size_t)b * HH + h) * NN + row;
      rowmax[idx] = runm[v];
      rowinv[idx] = 1.0f / ps;
    }
  }
}

// ---------------------------------------------------------------------------
// Kernel 3: out[d,j] = sum_i v[d,i] * exp(s[i,j]-m_i)/Z_i.
// One wave computes a 64(d) x 32(j) output tile; loops i in steps of 16:
//   (a) WMMA-recompute S tile (16 x 32), (b) normalize in registers,
//   (c) relayout P from D-layout to B-layout via __shfl, (d) WMMA V*P.
// ---------------------------------------------------------------------------
__global__ void k_attn_av(const float* __restrict__ qkv,
                          const float* __restrict__ rowmax,
                          const float* __restrict__ rowinv,
                          float* __restrict__ attout) {    // [BB][CC][NN]
  const int lane = threadIdx.x;
  const int nlo  = lane & 15;
  const int half = lane >> 4;
  const int kk   = half * 2;
  const int j0   = blockIdx.x * 32;
  const int h    = blockIdx.y;
  const int b    = blockIdx.z;

  const float* base = qkv + ((size_t)b * (3 * CC) + (size_t)h * (3 * HD)) * NN;
  const float* qp = base;
  const float* kp = base + (size_t)HD * NN;
  const float* vp = base + (size_t)(2 * HD) * NN;
  const float* mrow = rowmax + ((size_t)b * HH + h) * NN;
  const float* rrow = rowinv + ((size_t)b * HH + h) * NN;

  v8f o[4][2];
#pragma unroll
  for (int mt = 0; mt < 4; ++mt)
#pragma unroll
    for (int t = 0; t < 2; ++t) o[mt][t] = zero8();

  for (int i0 = 0; i0 < NN; i0 += 16) {
    // (a) S tile: P[i_local, j_local], i_local in [0,16), j_local in [0,32)
    v8f p[2];
    p[0] = zero8();
    p[1] = zero8();
#pragma unroll
    for (int ks = 0; ks < 16; ++ks) {
      const float* ap = kp + (size_t)(4 * ks + kk) * NN + i0 + nlo;
      v2f a;
      a.x = ap[0];
      a.y = ap[NN];
#pragma unroll
      for (int t = 0; t < 2; ++t) {
        const float* qpp = qp + (size_t)(4 * ks + kk) * NN + j0 + t * 16 + nlo;
        v2f bv;
        bv.x = qpp[0];
        bv.y = qpp[NN];
        p[t] = wmma_f32(a, bv, p[t]);
      }
    }
    // (b) normalize: element v holds row i0 + v + 8*half
#pragma unroll
    for (int v = 0; v < 8; ++v) {
      const int row = i0 + v + 8 * half;
      const float m = mrow[row];
      const float r = rrow[row];
#pragma unroll
      for (int t = 0; t < 2; ++t)
        p[t][v] = __expf(p[t][v] * SCALE_F - m) * r;
    }
    // (c)+(d) out += V(64x16) * P(16x32), contraction in 4 k-steps.
    // B operand rows r0=4ks+2*half_dest, r1=r0+1 live in D-layout regs of
    // source lane nlo + 16*(ks>>1); element index depends on dest half, so
    // shuffle both candidates and select.
#pragma unroll
    for (int ks = 0; ks < 4; ++ks) {
      const int srcLane = nlo + ((ks >> 1) << 4);
      v2f bv[2];
#pragma unroll
      for (int t = 0; t < 2; ++t) {
        const float bx_lo = __shfl(p[t][(4 * ks) & 7], srcLane, 32);
        const float bx_hi = __shfl(p[t][(4 * ks + 2) & 7], srcLane, 32);
        const float by_lo = __shfl(p[t][(4 * ks + 1) & 7], srcLane, 32);
        const float by_hi = __shfl(p[t][(4 * ks + 3) & 7], srcLane, 32);
        bv[t].x = half ? bx_hi : bx_lo;
        bv[t].y = half ? by_hi : by_lo;
      }
#pragma unroll
      for (int mt = 0; mt < 4; ++mt) {
        const float* apv = vp + (size_t)(mt * 16 + nlo) * NN + i0 + 4 * ks + kk;
        v2f a;
        a.x = apv[0];
        a.y = apv[1];
#pragma unroll
        for (int t = 0; t < 2; ++t) o[mt][t] = wmma_f32(a, bv[t], o[mt][t]);
      }
    }
  }

  // Store: channel c = h*HD + d, d = mt*16 + v + 8*half
#pragma unroll
  for (int mt = 0; mt < 4; ++mt)
#pragma unroll
    for (int v = 0; v < 8; ++v) {
      const int d = mt * 16 + v + 8 * half;
      float* yp = attout + ((size_t)b * CC + h * HD + d) * NN + j0 + nlo;
      yp[0]  = o[mt][0][v];
      yp[16] = o[mt][1][v];
    }
}

// ---------------------------------------------------------------------------
extern "C" void kernel_launch(void* const* d_in, const int* in_sizes, int n_in,
                              void* d_out, int out_size, void* d_ws, size_t ws_size,
                              hipStream_t stream) {
  const float* x      = (const float*)d_in[0];  // [4,512,2048]
  const float* w_qkv  = (const float*)d_in[1];  // [1536,512]
  const float* w_proj = (const float*)d_in[2];  // [512,512]
  const float* b_proj = (const float*)d_in[3];  // [512]
  float* out = (float*)d_out;                   // [4,512,2048]

  float* qkv    = (float*)d_ws;                          // 12.58M floats (48 MB)
  float* rowmax = qkv + (size_t)BB * 3 * CC * NN;        // 64K floats
  float* rowinv = rowmax + (size_t)BB * HH * NN;         // 64K floats
  float* attout = rowinv + (size_t)BB * HH * NN;         // 4.19M floats (16 MB)

  const dim3 blk(32);
  k_gemm_wn<<<dim3(NN / 64, (3 * CC) / 32, BB), blk, 0, stream>>>(
      w_qkv, x, qkv, nullptr, 3 * CC);
  k_attn_stats<<<dim3(NN / 16, HH, BB), blk, 0, stream>>>(qkv, rowmax, rowinv);
  k_attn_av<<<dim3(NN / 32, HH, BB), blk, 0, stream>>>(qkv, rowmax, rowinv, attout);
  k_gemm_wn<<<dim3(NN / 64, CC / 32, BB), blk, 0, stream>>>(
      w_proj, attout, out, b_proj, CC);
}